// EAFNN_25323127177827
// MI455X (gfx1250) — compile-verified
//
#include <hip/hip_runtime.h>
#include <hip/hip_bf16.h>

typedef float v2f __attribute__((ext_vector_type(2)));
typedef float v8f __attribute__((ext_vector_type(8)));

#define Nn 500000
#define Ff 8
#define Ll 16
#define Xn 8
#define Yn 4
#define NT (Nn / 16)
#define OUT_OFF ((size_t)Ll * (size_t)Nn)

#define WAVES 8
#define LDS_PER_WAVE 448  // 128 inp + 256 fir + 64 out (floats)

__global__ __launch_bounds__(256) void eafnn_kernel(
    const float* __restrict__ inp, const float* __restrict__ x,
    const float* __restrict__ c, const float* __restrict__ b,
    const float* __restrict__ w, float* __restrict__ out)
{
    __shared__ float smem[WAVES * LDS_PER_WAVE];

    const int lane  = threadIdx.x & 31;
    const int laneN = lane & 15;   // rule index l in the firing stage
    const int h     = lane >> 4;   // half of the wave
    const int waveInBlock = threadIdx.x >> 5;

    float* lds_inp = smem + waveInBlock * LDS_PER_WAVE;  // [16 n][8 f]
    float* lds_fir = lds_inp + 128;                      // [16 l][16 n], raw firings
    float* lds_out = lds_fir + 256;                      // [4 y][16 n]

    // ---- hoisted per-lane constants: centers / -1/(2b^2) for rule l = laneN ----
    float cc[Ff], sb[Ff];
#pragma unroll
    for (int f = 0; f < Ff; ++f) {
        cc[f] = c[f * Ll + laneN];
        float bb = b[f * Ll + laneN];
        sb[f] = -0.5f / (bb * bb);
    }

    // ---- hoisted WMMA B operands: W2[k][col] = w[l, k, y], col = 16t+laneN ----
    // B 4x16 f32 layout (mirror of documented A 16x4): VGPR j holds k = j + 2h.
    v2f Bm[2][4];
#pragma unroll
    for (int s = 0; s < 2; ++s) {
#pragma unroll
        for (int t = 0; t < 4; ++t) {
            int col = 16 * t + laneN;
            int l = col >> 2, y = col & 3;
#pragma unroll
            for (int j = 0; j < 2; ++j) {
                int k = 4 * s + j + 2 * h;
                Bm[s][t][j] = w[l * (Xn * Yn) + k * Yn + y];
            }
        }
    }

    const int gwave  = (int)((blockIdx.x * blockDim.x + threadIdx.x) >> 5);
    const int nwaves = (int)((gridDim.x * blockDim.x) >> 5);

    for (int tile = gwave; tile < NT; tile += nwaves) {
        const int n0 = tile * 16;

        // speculative prefetch of the next tile (global_prefetch_b8)
        __builtin_prefetch(inp + (size_t)(tile + nwaves) * 128, 0, 1);
        __builtin_prefetch(x   + (size_t)(tile + nwaves) * 128, 0, 1);

        // ---- stage inp tile [16n x 8f] into LDS via async DMA (ASYNCcnt path) ----
        {
            unsigned ldsoff = (unsigned)(uintptr_t)(lds_inp + lane * 4);
            unsigned long long ga =
                (unsigned long long)(uintptr_t)(inp + (size_t)tile * 128 + lane * 4);
            asm volatile("global_load_async_to_lds_b128 %0, %1, off"
                         :: "v"(ldsoff), "v"(ga) : "memory");
        }

        // ---- WMMA A operands from x: lane(h,m): M=m, K = 4s + j + 2h ----
        v2f A0, A1;
        {
            const float* xr = x + (size_t)(n0 + laneN) * Xn;
            float2 p0 = *(const float2*)(xr + 2 * h);
            float2 p1 = *(const float2*)(xr + 4 + 2 * h);
            A0[0] = p0.x; A0[1] = p0.y;
            A1[0] = p1.x; A1[1] = p1.y;
        }

        asm volatile("s_wait_asynccnt 0x0" ::: "memory");

        // ---- firing strengths, one exp per (n,l): lane(h,l) covers n = n0+r+8h ----
        float fir[8];
#pragma unroll
        for (int r = 0; r < 8; ++r) {
            const float* row = lds_inp + (r + 8 * h) * Ff;
            float4 q0 = *(const float4*)(row);
            float4 q1 = *(const float4*)(row + 4);
            float p[8] = {q0.x, q0.y, q0.z, q0.w, q1.x, q1.y, q1.z, q1.w};
            float e = 0.0f;
#pragma unroll
            for (int f = 0; f < Ff; ++f) {
                float d = p[f] - cc[f];
                e = fmaf(d * d, sb[f], e);
            }
            fir[r] = __expf(e) + 0.001f;
        }

        // ---- raw firings into LDS, transposed layout [l][m] ----
        {
            float4 q0 = {fir[0], fir[1], fir[2], fir[3]};
            float4 q1 = {fir[4], fir[5], fir[6], fir[7]};
            *(float4*)(lds_fir + laneN * 16 + 8 * h)     = q0;
            *(float4*)(lds_fir + laneN * 16 + 8 * h + 4) = q1;
        }
        asm volatile("s_wait_dscnt 0" ::: "memory");

        // ---- per-n total firing via LDS column sum; one rcp per lane ----
        // lane's n = n0 + laneN serves both the norm.T and out.T store stages.
        float tot = 0.0f;
#pragma unroll
        for (int l = 0; l < Ll; ++l) tot += lds_fir[l * 16 + laneN];
        const float rtot = __builtin_amdgcn_rcpf(tot);

        // ---- norm.T output: scale raw firings on the way out (coalesced) ----
#pragma unroll
        for (int jj = 0; jj < 8; ++jj) {
            int lrow = 2 * jj + h;
            out[(size_t)lrow * Nn + n0 + laneN] = lds_fir[lrow * 16 + laneN] * rtot;
        }

        // ---- rule_out GEMM on the matrix core: [16n x 8k] @ [8k x 64(l,y)] ----
        v8f acc[4];
#pragma unroll
        for (int t = 0; t < 4; ++t) {
            v8f z = {};
            acc[t] = __builtin_amdgcn_wmma_f32_16x16x4_f32(
                false, A0, false, Bm[0][t], (short)0, z, false, false);
            acc[t] = __builtin_amdgcn_wmma_f32_16x16x4_f32(
                false, A1, false, Bm[1][t], (short)0, acc[t], false, false);
        }

        // ---- weighted combine with RAW firings (normalize at the end) ----
        // acc[t][r] holds R[n0+r+8h, l=4t+(laneN>>2), y=laneN&3]; the firing
        // weights sit contiguously in lds_fir[l*16+8h ...] -> 2x b128 per tile.
        float so[8];
#pragma unroll
        for (int r = 0; r < 8; ++r) so[r] = 0.0f;
#pragma unroll
        for (int t = 0; t < 4; ++t) {
            const float* src = lds_fir + (4 * t + (laneN >> 2)) * 16 + 8 * h;
            float4 g0 = *(const float4*)(src);
            float4 g1 = *(const float4*)(src + 4);
            so[0] = fmaf(g0.x, acc[t][0], so[0]);
            so[1] = fmaf(g0.y, acc[t][1], so[1]);
            so[2] = fmaf(g0.z, acc[t][2], so[2]);
            so[3] = fmaf(g0.w, acc[t][3], so[3]);
            so[4] = fmaf(g1.x, acc[t][4], so[4]);
            so[5] = fmaf(g1.y, acc[t][5], so[5]);
            so[6] = fmaf(g1.z, acc[t][6], so[6]);
            so[7] = fmaf(g1.w, acc[t][7], so[7]);
        }
#pragma unroll
        for (int r = 0; r < 8; ++r) {   // fold the remaining l-groups
            so[r] += __shfl_xor(so[r], 4, 32);
            so[r] += __shfl_xor(so[r], 8, 32);
        }

        // ---- out.T output: transpose through LDS, divide by total at the store ----
        if ((laneN >> 2) == 0) {
            int y = laneN & 3;
            float4 q0 = {so[0], so[1], so[2], so[3]};
            float4 q1 = {so[4], so[5], so[6], so[7]};
            *(float4*)(lds_out + y * 16 + 8 * h)     = q0;
            *(float4*)(lds_out + y * 16 + 8 * h + 4) = q1;
        }
        asm volatile("s_wait_dscnt 0" ::: "memory");
        {
            int y0 = lane >> 4;
            int m0 = lane & 15;  // same laneN -> reuse rtot
            out[OUT_OFF + (size_t)y0 * Nn + n0 + m0]       = lds_out[lane]      * rtot;
            out[OUT_OFF + (size_t)(y0 + 2) * Nn + n0 + m0] = lds_out[lane + 32] * rtot;
        }
    }
}

extern "C" void kernel_launch(void* const* d_in, const int* in_sizes, int n_in,
                              void* d_out, int out_size, void* d_ws, size_t ws_size,
                              hipStream_t stream) {
    (void)in_sizes; (void)n_in; (void)out_size; (void)d_ws; (void)ws_size;
    const float* inp = (const float*)d_in[0];
    const float* x   = (const float*)d_in[1];
    const float* c   = (const float*)d_in[2];
    const float* b   = (const float*)d_in[3];
    const float* w   = (const float*)d_in[4];
    float* out = (float*)d_out;
    // 768 blocks x 256 threads = 6144 waves over 31250 tiles (~5 tiles/wave,
    // amortizing the hoisted B-matrix / center / width register setup).
    eafnn_kernel<<<dim3(768), dim3(256), 0, stream>>>(inp, x, c, b, w, out);
}